// FullyConnectedTensorProduct_84318797955726
// MI455X (gfx1250) — compile-verified
//
#include <hip/hip_runtime.h>

typedef __attribute__((ext_vector_type(2))) float v2f;
typedef __attribute__((ext_vector_type(8))) float v8f;

#define SAMPLES_PER_BLOCK 64   // 4 waves * 16 samples
#define X1ROW 132              // padded x1 row stride (floats): bank step 4 -> conflict-free
#define WROW  40               // padded weight row stride: lane halves hit disjoint bank groups
#define B1ROW 36               // padded b1 row stride
#define INV_SQRT3 0.5773502691896258f

__launch_bounds__(128)
__global__ void ftp_wmma_kernel(const float* __restrict__ x1,
                                const float* __restrict__ x2,
                                const float* __restrict__ w000,
                                const float* __restrict__ w011,
                                const float* __restrict__ w101,
                                const float* __restrict__ w110,
                                float* __restrict__ out, int n) {
    __shared__ float sW000[32 * WROW];
    __shared__ float sW011[32 * WROW];
    __shared__ float sW101[32 * WROW];
    __shared__ float sW110[32 * WROW];
    __shared__ float sX1[SAMPLES_PER_BLOCK * X1ROW];
    __shared__ float sX2[SAMPLES_PER_BLOCK * 4];
    __shared__ float sB1[4][16 * B1ROW];

    const int tid  = threadIdx.x;
    const int wave = tid >> 5;
    const int lane = tid & 31;
    const int base = blockIdx.x * SAMPLES_PER_BLOCK;

    // ---- stage weights (each 32x32 = 256 float4; 2 per thread per matrix) ----
    #pragma unroll
    for (int i = 0; i < 2; ++i) {
        int q = i * 128 + tid;          // float4 index within a matrix
        int u = q >> 3;                 // row
        int c = (q & 7) * 4;            // col
        float4 a = reinterpret_cast<const float4*>(w000)[q];
        float4 b = reinterpret_cast<const float4*>(w011)[q];
        float4 cc = reinterpret_cast<const float4*>(w101)[q];
        float4 d = reinterpret_cast<const float4*>(w110)[q];
        *reinterpret_cast<float4*>(&sW000[u * WROW + c]) = a;
        *reinterpret_cast<float4*>(&sW011[u * WROW + c]) = b;
        *reinterpret_cast<float4*>(&sW101[u * WROW + c]) = cc;
        *reinterpret_cast<float4*>(&sW110[u * WROW + c]) = d;
    }

    // ---- stage x1 tile: 64 rows x 128 floats = 2048 float4, coalesced ----
    #pragma unroll
    for (int i = 0; i < 16; ++i) {
        int idx  = i * 128 + tid;       // float4 index
        int row  = idx >> 5;            // 32 float4 per row
        int col4 = (idx & 31) * 4;
        int g    = base + row;
        float4 v = make_float4(0.f, 0.f, 0.f, 0.f);
        if (g < n) v = reinterpret_cast<const float4*>(x1)[(size_t)g * 32 + (idx & 31)];
        *reinterpret_cast<float4*>(&sX1[row * X1ROW + col4]) = v;
    }

    // ---- stage x2 tile: 64 rows x 1 float4 ----
    if (tid < SAMPLES_PER_BLOCK) {
        int g = base + tid;
        float4 v = make_float4(0.f, 0.f, 0.f, 0.f);
        if (g < n) v = reinterpret_cast<const float4*>(x2)[g];
        *reinterpret_cast<float4*>(&sX2[tid * 4]) = v;
    }
    __syncthreads();

    // ---- per-wave b1 tile: b1[m][u] = dot(x1_1[m,u,:], x2_1[m,:]) ----
    #pragma unroll
    for (int e = 0; e < 16; ++e) {
        int m = wave * 16 + e;
        const float* xr  = &sX1[m * X1ROW + 32 + 3 * lane];
        const float* x2r = &sX2[m * 4];
        sB1[wave][e * B1ROW + lane] = xr[0] * x2r[1] + xr[1] * x2r[2] + xr[2] * x2r[3];
    }
    __syncthreads();

    // ---- WMMA GEMMs: A = per-sample vectors (16x32), B = weights (32x32) ----
    const int mloc  = lane & 15;          // A row / C col index
    const int khalf = (lane >> 4) * 2;    // K offset of this lane half
    const float* myX1 = &sX1[(wave * 16 + mloc) * X1ROW];
    const float* myB1 = &sB1[wave][mloc * B1ROW];

    #pragma unroll
    for (int nb = 0; nb < 2; ++nb) {
        v8f accP = {}, accQ = {}, accR = {}, accS0 = {}, accS1 = {}, accS2 = {};
        #pragma unroll
        for (int kb = 0; kb < 8; ++kb) {
            const int k0 = kb * 4 + khalf;
            // A fragments (16x4 fp32: lanes 0-15 hold K=k0,k0+1; 16-31 hold K=k0 with khalf=2)
            v2f a0;  a0.x = myX1[k0];     a0.y = myX1[k0 + 1];        // x1_0
            v2f ab;  ab.x = myB1[k0];     ab.y = myB1[k0 + 1];        // b1
            const float* p0 = myX1 + 32 + 3 * k0;                     // x1_1[:,u=k0..k0+1,k]
            v2f as0; as0.x = p0[0]; as0.y = p0[3];
            v2f as1; as1.x = p0[1]; as1.y = p0[4];
            v2f as2; as2.x = p0[2]; as2.y = p0[5];
            // B fragments (4x16 fp32: VGPR0 row k0, VGPR1 row k0+1; col = lane&15)
            const int wr = k0 * WROW + nb * 16 + mloc;
            v2f b000; b000.x = sW000[wr]; b000.y = sW000[wr + WROW];
            v2f b011; b011.x = sW011[wr]; b011.y = sW011[wr + WROW];
            v2f b101; b101.x = sW101[wr]; b101.y = sW101[wr + WROW];
            v2f b110; b110.x = sW110[wr]; b110.y = sW110[wr + WROW];

            accP  = __builtin_amdgcn_wmma_f32_16x16x4_f32(false, a0,  false, b000, (short)0, accP,  false, false);
            accR  = __builtin_amdgcn_wmma_f32_16x16x4_f32(false, a0,  false, b011, (short)0, accR,  false, false);
            accQ  = __builtin_amdgcn_wmma_f32_16x16x4_f32(false, ab,  false, b110, (short)0, accQ,  false, false);
            accS0 = __builtin_amdgcn_wmma_f32_16x16x4_f32(false, as0, false, b101, (short)0, accS0, false, false);
            accS1 = __builtin_amdgcn_wmma_f32_16x16x4_f32(false, as1, false, b101, (short)0, accS1, false, false);
            accS2 = __builtin_amdgcn_wmma_f32_16x16x4_f32(false, as2, false, b101, (short)0, accS2, false, false);
        }

        // ---- epilogue: C/D layout reg v -> row v + 8*(lane>=16), col lane&15 ----
        #pragma unroll
        for (int v = 0; v < 8; ++v) {
            const int mrow = v + 8 * (lane >> 4);
            const int g    = base + wave * 16 + mrow;
            const float* x2r = &sX2[(wave * 16 + mrow) * 4];
            const float x20 = x2r[0];
            const int w = nb * 16 + mloc;
            const float o0  = x20 * accP[v] + INV_SQRT3 * accQ[v];
            const float o1a = x2r[1] * accR[v] + x20 * accS0[v];
            const float o1b = x2r[2] * accR[v] + x20 * accS1[v];
            const float o1c = x2r[3] * accR[v] + x20 * accS2[v];
            if (g < n) {
                float* orow = out + (size_t)g * 128;
                orow[w]              = o0;
                orow[32 + 3 * w + 0] = o1a;
                orow[32 + 3 * w + 1] = o1b;
                orow[32 + 3 * w + 2] = o1c;
            }
        }
    }
}

extern "C" void kernel_launch(void* const* d_in, const int* in_sizes, int n_in,
                              void* d_out, int out_size, void* d_ws, size_t ws_size,
                              hipStream_t stream) {
    const float* x1   = (const float*)d_in[0];
    const float* x2   = (const float*)d_in[1];
    const float* w000 = (const float*)d_in[2];
    const float* w011 = (const float*)d_in[3];
    const float* w101 = (const float*)d_in[4];
    const float* w110 = (const float*)d_in[5];
    float* out = (float*)d_out;
    const int n = in_sizes[0] / 128;
    const int grid = (n + SAMPLES_PER_BLOCK - 1) / SAMPLES_PER_BLOCK;
    ftp_wmma_kernel<<<grid, 128, 0, stream>>>(x1, x2, w000, w011, w101, w110, out, n);
}